// PolyphonicDDSPPianoDynamic_32014686224485
// MI455X (gfx1250) — compile-verified
//
#include <hip/hip_runtime.h>
#include <math.h>

// ---------------------------------------------------------------------------
// PolyphonicDDSPPiano forward for MI455X (gfx1250, wave32).
// Batched GEMMs use V_WMMA_F32_16X16X4_F32 (fp32 matrix path, exactness
// matters: f_k feeds a 64000-sample phase cumsum). Each wave owns a 16x32
// output tile (2 accumulators share one A fragment); all K dims padded to a
// multiple of 4 so the inner loop is guard-free (plain b64/b32 loads, EXEC
// stays all-ones for WMMA). Sequential GRU scans are VALU (latency chains,
// weights L2/LDS resident). Harmonic synth = block-scan cumsum + atomic f32
// accumulation. Reverb = direct linear convolution (identical to the
// reference's zero-padded FFT product).
// ---------------------------------------------------------------------------

typedef float v2f __attribute__((ext_vector_type(2)));
typedef float v8f __attribute__((ext_vector_type(8)));

#define HAVE_WMMA_F32 0
#if defined(__has_builtin)
#if __has_builtin(__builtin_amdgcn_wmma_f32_16x16x4_f32)
#undef HAVE_WMMA_F32
#define HAVE_WMMA_F32 1
#endif
#endif

#define B_   2
#define T_   1000
#define HOP_ 64
#define WIN_ 128
#define NH_  96
#define NNF_ 64
#define CTX_ 32
#define CU_  64
#define HID_ 512
#define U_   512
#define NS_  64000
#define IRL_ 24000
#define ROWS_ (B_ * T_)   // 2000, divisible by 16
#define KIN_  36          // core_in K padded 34 -> 36 (multiple of 4)

__device__ __forceinline__ float dsigmoid(float x) { return 1.0f / (1.0f + expf(-x)); }
__device__ __forceinline__ float dleaky(float x)   { return x > 0.0f ? x : 0.2f * x; }
__device__ __forceinline__ float dmodsig(float x) {
  float s = dsigmoid(x);
  return 2.0f * powf(s, 2.3025850929940457f) + 1e-7f;  // 2*sig(x)^ln(10)+1e-7
}

// ------------------------- generic WMMA fp32 GEMM --------------------------
// C[M,N] = act(A[M,K] @ W[K,N] + bias[N]); one 16x32 tile per wave.
// Requirements: M%16==0, N%32==0, K%4==0, A/W 8-byte aligned.
__global__ void k_wmma_gemm(const float* __restrict__ A, const float* __restrict__ W,
                            const float* __restrict__ bias, float* __restrict__ C,
                            int M, int N, int K, int act) {
  int wave = threadIdx.x >> 5;
  int lane = threadIdx.x & 31;
  int tile = blockIdx.x * 8 + wave;
  int ntn  = N >> 5;                    // 32-wide column tiles
  int ntiles = (M >> 4) * ntn;
  if (tile >= ntiles) return;           // wave-uniform: EXEC stays all-ones
  int m0 = (tile / ntn) << 4;
  int n0 = (tile % ntn) << 5;
  int lm = lane & 15;
  int half = lane >> 4;
  // A 16x4 frag (ISA 7.12.2): lanes0-15 hold {K=k0,k0+1}, lanes16-31 {k0+2,k0+3}
  const float* arow = A + (size_t)(m0 + lm) * K + half * 2;
  // B 4x16 frag: VGPR0 = rows {k0 | k0+2}, VGPR1 = rows {k0+1 | k0+3}
  const float* wcol = W + n0 + lm;
#if HAVE_WMMA_F32
  v8f acc0 = {}, acc1 = {};
#pragma unroll 4
  for (int k0 = 0; k0 < K; k0 += 4) {
    v2f a = *(const v2f*)(arow + k0);                 // one b64 load
    const float* wr0 = wcol + (size_t)(k0 + half * 2) * N;
    const float* wr1 = wr0 + N;
    v2f b0, b1;
    b0.x = wr0[0];  b0.y = wr1[0];
    b1.x = wr0[16]; b1.y = wr1[16];
    acc0 = __builtin_amdgcn_wmma_f32_16x16x4_f32(false, a, false, b0,
                                                 (short)0, acc0, false, false);
    acc1 = __builtin_amdgcn_wmma_f32_16x16x4_f32(false, a, false, b1,
                                                 (short)0, acc1, false, false);
  }
  for (int r = 0; r < 8; ++r) {
    int row = m0 + r + half * 8;   // C: VGPR r holds rows r (lanes0-15), r+8
    float v0 = acc0[r] + bias[n0 + lm];
    float v1 = acc1[r] + bias[n0 + 16 + lm];
    if (act == 1) { v0 = dleaky(v0); v1 = dleaky(v1); }
    C[(size_t)row * N + n0 + lm]      = v0;
    C[(size_t)row * N + n0 + 16 + lm] = v1;
  }
#else
  for (int r = 0; r < 8; ++r) {
    int row = m0 + r + half * 8;
    for (int c = 0; c < 2; ++c) {
      int col = n0 + c * 16 + lm;
      float s = 0.0f;
      for (int k = 0; k < K; ++k) s += A[(size_t)row * K + k] * W[(size_t)k * N + col];
      float v = s + bias[col];
      if (act == 1) v = dleaky(v);
      C[(size_t)row * N + col] = v;
    }
  }
#endif
}

// ---------------------- context net front (ctx_in -> c0) -------------------
__global__ void k_ctx_front(const float* __restrict__ pitches, const float* __restrict__ vels,
                            const float* __restrict__ pedal, const float* __restrict__ Wi,
                            const float* __restrict__ bi, float* __restrict__ c0) {
  int row = blockIdx.x;          // b*T + t  (V==1 so pitches/vels flatten to row)
  int j = threadIdx.x;           // 0..31
  float p = pitches[row], v = vels[row], ped = pedal[row];
  float gvel = v;                                        // sum over V=1
  float act = (v > 0.0f) ? p / 127.0f : 0.0f;
  float cnt = (v > 0.0f) ? 1.0f : 0.0f;
  float gp = act / (cnt + 1e-7f);
  float s = ped * Wi[0 * CTX_ + j] + gvel * Wi[1 * CTX_ + j] + gp * Wi[2 * CTX_ + j] + bi[j];
  c0[row * CTX_ + j] = dleaky(s);
}

// -------------------------- ctx GRU scan (CU=64) ---------------------------
__global__ void k_ctx_scan(const float* __restrict__ xw, const float* __restrict__ rk,
                           const float* __restrict__ gb1, float* __restrict__ hs) {
  __shared__ float s_rk[CU_ * 3 * CU_];   // 48KB, fits LDS
  __shared__ float s_h[CU_];
  int b = blockIdx.x, u = threadIdx.x;
  for (int i = u; i < CU_ * 3 * CU_; i += CU_) s_rk[i] = rk[i];
  s_h[u] = 0.0f;
  float bz = gb1[u], br = gb1[CU_ + u], bh = gb1[2 * CU_ + u];
  __syncthreads();
  for (int t = 0; t < T_; ++t) {
    const float* x = xw + (b * T_ + t) * 3 * CU_;
    float dz = 0.0f, dr = 0.0f, dh = 0.0f;
    for (int v = 0; v < CU_; ++v) {
      float hv = s_h[v];
      dz += hv * s_rk[v * 3 * CU_ + u];
      dr += hv * s_rk[v * 3 * CU_ + CU_ + u];
      dh += hv * s_rk[v * 3 * CU_ + 2 * CU_ + u];
    }
    float hold = s_h[u];
    float z  = dsigmoid(x[u] + dz + bz);
    float r  = dsigmoid(x[CU_ + u] + dr + br);
    float hh = tanhf(x[2 * CU_ + u] + r * (dh + bh));
    float hn = z * hold + (1.0f - z) * hh;
    __syncthreads();
    s_h[u] = hn;
    hs[(b * T_ + t) * CU_ + u] = hn;
    __syncthreads();
  }
}

// ------- assemble core_in = [pitch/127, vel, c_t, 0, 0] (K padded to 36) ----
__global__ void k_corein(const float* __restrict__ pitches, const float* __restrict__ vels,
                         const float* __restrict__ ct, float* __restrict__ ci, int n) {
  int i = blockIdx.x * blockDim.x + threadIdx.x;
  if (i >= n) return;
  int row = i / KIN_, col = i % KIN_;
  float v;
  if (col == 0)       v = pitches[row] / 127.0f;
  else if (col == 1)  v = vels[row];
  else if (col < 34)  v = ct[row * CTX_ + (col - 2)];
  else                v = 0.0f;                       // zero padding
  ci[i] = v;
}

// ----------- zero-pad core_Wi (34x512) into a 36x512 ws copy ---------------
__global__ void k_padWi(const float* __restrict__ Wi, float* __restrict__ Wp, int n) {
  int i = blockIdx.x * blockDim.x + threadIdx.x;
  if (i >= n) return;
  int row = i / HID_;
  Wp[i] = (row < 34) ? Wi[i] : 0.0f;
}

// -------------------------- core GRU scan (U=512) --------------------------
__global__ void k_core_scan(const float* __restrict__ xw, const float* __restrict__ rk,
                            const float* __restrict__ gb1, float* __restrict__ hs) {
  __shared__ float s_h[U_];
  __shared__ float s_pz[U_], s_pr[U_], s_ph[U_];
  int b = blockIdx.x;
  int tid = threadIdx.x;        // 0..1023
  int u = tid & (U_ - 1);
  int half = tid >> 9;          // split the 512-dot across two threads
  int v0 = half * 256, v1 = v0 + 256;
  if (half == 0) s_h[u] = 0.0f;
  __syncthreads();
  for (int t = 0; t < T_; ++t) {
    float dz = 0.0f, dr = 0.0f, dh = 0.0f;
    for (int v = v0; v < v1; ++v) {
      float hv = s_h[v];
      const float* w = rk + (size_t)v * 3 * U_ + u;   // coalesced across u
      dz += hv * w[0];
      dr += hv * w[U_];
      dh += hv * w[2 * U_];
    }
    float hold = s_h[u];
    __syncthreads();
    if (half == 1) { s_pz[u] = dz; s_pr[u] = dr; s_ph[u] = dh; }
    __syncthreads();
    if (half == 0) {
      dz += s_pz[u]; dr += s_pr[u]; dh += s_ph[u];
      const float* x = xw + (size_t)(b * T_ + t) * 3 * U_;
      float z  = dsigmoid(x[u] + dz + gb1[u]);
      float r  = dsigmoid(x[U_ + u] + dr + gb1[U_ + u]);
      float hh = tanhf(x[2 * U_ + u] + r * (dh + gb1[2 * U_ + u]));
      float hn = z * hold + (1.0f - z) * hh;
      s_h[u] = hn;
      hs[(size_t)(b * T_ + t) * U_ + u] = hn;
    }
    __syncthreads();
  }
}

// ------------------------------ amp head (N=1) -----------------------------
__global__ void k_amp(const float* __restrict__ h2, const float* __restrict__ ampW,
                      const float* __restrict__ ampb, float* __restrict__ ampraw) {
  int row = blockIdx.x * blockDim.x + threadIdx.x;
  if (row >= ROWS_) return;
  float s = ampb[0];
  for (int i = 0; i < HID_; ++i) s += h2[(size_t)row * HID_ + i] * ampW[i];
  ampraw[row] = s;
}

// ---------------- per-frame post: sigmoids, norm, f_k, f_k2 ----------------
__global__ void k_post(const float* __restrict__ pitches, const float* __restrict__ harmraw,
                       const float* __restrict__ noiseraw, const float* __restrict__ ampraw,
                       const float* __restrict__ inharm, const float* __restrict__ detW,
                       const float* __restrict__ detb, const float* __restrict__ dets,
                       float* __restrict__ amps, float* __restrict__ harms,
                       float* __restrict__ nmags, float* __restrict__ fk,
                       float* __restrict__ fk2) {
  __shared__ float red[128];
  int row = blockIdx.x, tid = threadIdx.x;  // 128 threads
  float hr = 0.0f;
  if (tid < NH_) hr = dmodsig(harmraw[row * NH_ + tid]);
  red[tid] = (tid < NH_) ? hr : 0.0f;
  __syncthreads();
  for (int off = 64; off > 0; off >>= 1) {
    if (tid < off) red[tid] += red[tid + off];
    __syncthreads();
  }
  float hsum = red[0];
  if (tid < NH_) harms[row * NH_ + tid] = hr / hsum;
  if (tid < NNF_) nmags[row * NNF_ + tid] = dmodsig(noiseraw[row * NNF_ + tid]);
  if (tid == 0) amps[row] = dmodsig(ampraw[row]);
  if (tid < NH_) {
    float vp = pitches[row];
    float f0 = (vp > 0.0f) ? 440.0f * exp2f((vp - 69.0f) / 12.0f) : 0.0f;
    float ps = (vp > 0.0f) ? vp : 1.0f;
    float aB = inharm[0], bB = inharm[1], aT = inharm[2], bT = inharm[3];
    float Bf = expf(aT * ps + bT) + expf(aB * ps + bB);
    float kh = (float)(tid + 1);
    float f = f0 * kh * sqrtf(1.0f + Bf * kh * kh);
    fk[row * NH_ + tid] = f;
    float delta = tanhf((vp / 127.0f) * detW[0] + detb[0]) + tanhf(dets[0]);
    fk2[row * NH_ + tid] = f * exp2f(delta / 12.0f);
  }
}

__global__ void k_zero(float* __restrict__ p, int n) {
  int i = blockIdx.x * blockDim.x + threadIdx.x;
  if (i < n) p[i] = 0.0f;
}

// ------------- harmonic synth: block-scan cumsum + atomic accum ------------
__global__ void k_harm(const float* __restrict__ fk, const float* __restrict__ fk2,
                       const float* __restrict__ harms, float* __restrict__ S) {
  __shared__ float sc[256];
  int blk = blockIdx.x;                 // b*192 + which*96 + h
  int b = blk / 192;
  int rem = blk % 192;
  int which = rem / NH_;
  int h = rem % NH_;
  const float* f = which ? fk2 : fk;
  int tid = threadIdx.x;
  float carry = 0.0f;
  const float w2pi = 2.0f * 3.14159265358979323846f / 16000.0f;
  for (int tile = 0; tile < NS_ / 256; ++tile) {
    int t = tile * 256 + tid;
    float pos = (t + 0.5f) / 64.0f - 0.5f;          // half-pixel bilinear
    int i0 = (int)floorf(pos);
    float w = pos - (float)i0;
    int ia = i0 < 0 ? 0 : (i0 > T_ - 1 ? T_ - 1 : i0);
    int i1 = i0 + 1;
    int ib = i1 < 0 ? 0 : (i1 > T_ - 1 ? T_ - 1 : i1);
    float fA = f[(b * T_ + ia) * NH_ + h], fB = f[(b * T_ + ib) * NH_ + h];
    float fup = fA + (fB - fA) * w;
    float val = fup * w2pi;
    sc[tid] = val;
    __syncthreads();
    for (int off = 1; off < 256; off <<= 1) {        // Hillis-Steele scan
      float tmp = (tid >= off) ? sc[tid - off] : 0.0f;
      __syncthreads();
      sc[tid] += tmp;
      __syncthreads();
    }
    float phase = carry + sc[tid];
    float total = sc[255];
    float hA = harms[(b * T_ + ia) * NH_ + h], hB = harms[(b * T_ + ib) * NH_ + h];
    float hup = hA + (hB - hA) * w;
    float mask = (fup < 8000.0f) ? 1.0f : 0.0f;
    atomicAdd(&S[b * NS_ + t], sinf(phase) * hup * mask);
    carry += total;
    __syncthreads();
  }
}

// ----- noise synth: circular conv with g = irfft(mags), window + OLA -------
__global__ void k_noise(const float* __restrict__ wn, const float* __restrict__ nmag,
                        float* __restrict__ nout) {
  __shared__ float fr[WIN_], g[WIN_], M[NNF_], ft[WIN_];
  int b = blockIdx.x / T_;
  int fr_i = blockIdx.x % T_;
  int tid = threadIdx.x;                // 0..127
  int idx = fr_i * HOP_ + tid;
  fr[tid] = (idx < NS_) ? wn[b * NS_ + idx] : 0.0f;   // zero-pad tail
  if (tid < NNF_) M[tid] = nmag[(b * T_ + fr_i) * NNF_ + tid];
  __syncthreads();
  // g[d] = (M0 + 2*sum_{k=1}^{63} Mk*cos(2pi*k*d/128)) / 128   (bin 64 == 0)
  float acc = M[0];
  for (int k = 1; k < NNF_; ++k)
    acc += 2.0f * M[k] * cosf(6.283185307179586f * (float)(k * tid) / 128.0f);
  g[tid] = acc / 128.0f;
  __syncthreads();
  float s = 0.0f;
  for (int m = 0; m < WIN_; ++m) s += fr[m] * g[(tid - m) & 127];   // circular conv
  ft[tid] = s;
  __syncthreads();
  float rolled = ft[(tid - 64) & 127];                              // roll by WIN/2
  float hann = 0.5f - 0.5f * cosf(6.283185307179586f * (float)tid / 128.0f);
  atomicAdd(&nout[b * (NS_ + HOP_) + fr_i * HOP_ + tid], rolled * hann);
}

// -------------- combine: audio = S*a_up + noise (trunc to NS) --------------
__global__ void k_combine(const float* __restrict__ S, const float* __restrict__ nout,
                          const float* __restrict__ amps, float* __restrict__ audio) {
  int i = blockIdx.x * blockDim.x + threadIdx.x;
  if (i >= B_ * NS_) return;
  int b = i / NS_, t = i % NS_;
  float pos = (t + 0.5f) / 64.0f - 0.5f;
  int i0 = (int)floorf(pos);
  float w = pos - (float)i0;
  int ia = i0 < 0 ? 0 : (i0 > T_ - 1 ? T_ - 1 : i0);
  int i1 = i0 + 1;
  int ib = i1 < 0 ? 0 : (i1 > T_ - 1 ? T_ - 1 : i1);
  float aA = amps[b * T_ + ia], aB = amps[b * T_ + ib];
  float a = aA + (aB - aA) * w;
  audio[i] = S[i] * a + nout[b * (NS_ + HOP_) + t];
}

// --------- reverb: LDS-tiled direct linear convolution (== FFT ref) --------
__global__ void k_reverb(const float* __restrict__ audio, const float* __restrict__ ir,
                         float* __restrict__ out) {
  __shared__ float at[512];
  __shared__ float irt[256];
  int blk = blockIdx.x;
  int b = blk / (NS_ / 256);
  int t0 = (blk % (NS_ / 256)) * 256;
  int tid = threadIdx.x;
  float acc = 0.0f;
  for (int c0 = 0; c0 < IRL_; c0 += 256) {
    irt[tid] = (c0 + tid < IRL_) ? ir[c0 + tid] : 0.0f;
    int base = t0 - c0 - 255;
    int a0 = base + tid, a1 = base + 256 + tid;
    at[tid]       = (a0 >= 0 && a0 < NS_) ? audio[b * NS_ + a0] : 0.0f;
    at[tid + 256] = (a1 >= 0 && a1 < NS_) ? audio[b * NS_ + a1] : 0.0f;
    __syncthreads();
#pragma unroll 8
    for (int j = 0; j < 256; ++j) acc += at[tid + 255 - j] * irt[j];
    __syncthreads();
  }
  out[b * NS_ + t0 + tid] = acc;
}

// ---------------------------------------------------------------------------
extern "C" void kernel_launch(void* const* d_in, const int* in_sizes, int n_in,
                              void* d_out, int out_size, void* d_ws, size_t ws_size,
                              hipStream_t stream) {
  (void)in_sizes; (void)n_in; (void)out_size; (void)ws_size;
  const float* pitches  = (const float*)d_in[0];
  const float* vels     = (const float*)d_in[1];
  const float* pedal    = (const float*)d_in[2];
  const float* wn       = (const float*)d_in[3];
  const float* ctx_Wi   = (const float*)d_in[4];
  const float* ctx_bi   = (const float*)d_in[5];
  const float* ctx_gk   = (const float*)d_in[6];
  const float* ctx_grk  = (const float*)d_in[7];
  const float* ctx_gb   = (const float*)d_in[8];
  const float* ctx_Wo   = (const float*)d_in[9];
  const float* ctx_bo   = (const float*)d_in[10];
  const float* core_Wi  = (const float*)d_in[11];
  const float* core_bi  = (const float*)d_in[12];
  const float* core_gk  = (const float*)d_in[13];
  const float* core_grk = (const float*)d_in[14];
  const float* core_gb  = (const float*)d_in[15];
  const float* core_Wh  = (const float*)d_in[16];
  const float* core_bh  = (const float*)d_in[17];
  const float* amp_W    = (const float*)d_in[18];
  const float* amp_b    = (const float*)d_in[19];
  const float* harm_W   = (const float*)d_in[20];
  const float* harm_b   = (const float*)d_in[21];
  const float* noise_W  = (const float*)d_in[22];
  const float* noise_b  = (const float*)d_in[23];
  const float* inharm   = (const float*)d_in[24];
  const float* det_W    = (const float*)d_in[25];
  const float* det_b    = (const float*)d_in[26];
  const float* det_s    = (const float*)d_in[27];
  const float* rev_ir   = (const float*)d_in[28];
  float* out = (float*)d_out;

  float* ws = (float*)d_ws;
  size_t off = 0;
  auto alloc = [&](size_t n) { float* p = ws + off; off += (n + 255) & ~(size_t)255; return p; };
  float* c0       = alloc(ROWS_ * CTX_);
  float* xwctx    = alloc(ROWS_ * 3 * CU_);
  float* ctxh     = alloc(ROWS_ * CU_);
  float* ct       = alloc(ROWS_ * CTX_);
  float* corein   = alloc(ROWS_ * KIN_);
  float* Wip      = alloc(KIN_ * HID_);              // padded core_Wi
  float* hin      = alloc((size_t)ROWS_ * HID_);
  float* xwcore   = alloc((size_t)ROWS_ * 3 * U_);
  float* hgru     = alloc((size_t)ROWS_ * U_);
  float* h2       = alloc((size_t)ROWS_ * HID_);
  float* harmraw  = alloc(ROWS_ * NH_);
  float* noiseraw = alloc(ROWS_ * NNF_);
  float* ampraw   = alloc(ROWS_);
  float* amps     = alloc(ROWS_);
  float* harms    = alloc(ROWS_ * NH_);
  float* nmags    = alloc(ROWS_ * NNF_);
  float* fk       = alloc(ROWS_ * NH_);
  float* fk2      = alloc(ROWS_ * NH_);
  float* S        = alloc(B_ * NS_);
  float* nout     = alloc(B_ * (NS_ + HOP_));
  float* audio    = alloc(B_ * NS_);

  auto gemm = [&](const float* A, const float* W, const float* bias, float* C,
                  int M, int N, int K, int act) {
    int tiles = (M / 16) * (N / 32);
    int blocks = (tiles + 7) / 8;
    k_wmma_gemm<<<blocks, 256, 0, stream>>>(A, W, bias, C, M, N, K, act);
  };

  // context net
  k_ctx_front<<<ROWS_, 32, 0, stream>>>(pitches, vels, pedal, ctx_Wi, ctx_bi, c0);
  gemm(c0, ctx_gk, ctx_gb, xwctx, ROWS_, 3 * CU_, CTX_, 0);          // + gb[0]
  k_ctx_scan<<<B_, CU_, 0, stream>>>(xwctx, ctx_grk, ctx_gb + 3 * CU_, ctxh);
  gemm(ctxh, ctx_Wo, ctx_bo, ct, ROWS_, CTX_, CU_, 0);

  // core net (K padded 34 -> 36 on both A and W so the GEMM loop is guard-free)
  k_corein<<<(ROWS_ * KIN_ + 255) / 256, 256, 0, stream>>>(pitches, vels, ct, corein, ROWS_ * KIN_);
  k_padWi<<<(KIN_ * HID_ + 255) / 256, 256, 0, stream>>>(core_Wi, Wip, KIN_ * HID_);
  gemm(corein, Wip, core_bi, hin, ROWS_, HID_, KIN_, 1);             // leaky 0.2
  gemm(hin, core_gk, core_gb, xwcore, ROWS_, 3 * U_, HID_, 0);       // + gb[0]
  k_core_scan<<<B_, 1024, 0, stream>>>(xwcore, core_grk, core_gb + 3 * U_, hgru);
  gemm(hgru, core_Wh, core_bh, h2, ROWS_, HID_, U_, 1);              // leaky 0.2

  // heads
  gemm(h2, harm_W, harm_b, harmraw, ROWS_, NH_, HID_, 0);
  gemm(h2, noise_W, noise_b, noiseraw, ROWS_, NNF_, HID_, 0);
  k_amp<<<(ROWS_ + 255) / 256, 256, 0, stream>>>(h2, amp_W, amp_b, ampraw);
  k_post<<<ROWS_, 128, 0, stream>>>(pitches, harmraw, noiseraw, ampraw, inharm,
                                    det_W, det_b, det_s, amps, harms, nmags, fk, fk2);

  // synthesis
  k_zero<<<(B_ * NS_ + 255) / 256, 256, 0, stream>>>(S, B_ * NS_);
  k_zero<<<(B_ * (NS_ + HOP_) + 255) / 256, 256, 0, stream>>>(nout, B_ * (NS_ + HOP_));
  k_harm<<<B_ * 2 * NH_, 256, 0, stream>>>(fk, fk2, harms, S);
  k_noise<<<B_ * T_, WIN_, 0, stream>>>(wn, nmags, nout);
  k_combine<<<(B_ * NS_ + 255) / 256, 256, 0, stream>>>(S, nout, amps, audio);

  // reverb -> d_out
  k_reverb<<<B_ * (NS_ / 256), 256, 0, stream>>>(audio, rev_ir, out);
}